// MambaVisionMixer_13065290515069
// MI455X (gfx1250) — compile-verified
//
#include <hip/hip_runtime.h>
#include <math.h>
#include <stdint.h>

// Problem dims (fixed by the reference)
#define B_    4
#define L_    1024
#define DM    1024   // d_model
#define DI    2048   // d_inner
#define DH    1024   // d_half
#define DS    16     // d_state
#define DTR   64     // dt_rank
#define NPROJ 96     // dt_rank + 2*d_state

typedef __attribute__((ext_vector_type(2))) float        v2f;
typedef __attribute__((ext_vector_type(8))) float        v8f;
typedef __attribute__((ext_vector_type(4))) unsigned int v4u;
typedef __attribute__((ext_vector_type(4))) int          v4i;
typedef __attribute__((ext_vector_type(8))) int          v8i;

__device__ __forceinline__ float softplus_f(float x) {
    return (x > 20.0f) ? x : log1pf(__expf(x));
}
__device__ __forceinline__ float gelu_f(float x) {
    return 0.5f * x * (1.0f + erff(x * 0.70710678118654752f));
}

// ---------------------------------------------------------------------------
// Tensor Data Mover: async 2-D tile load Global -> LDS with LDS padding.
// tile_w elements contiguous in x, tile_h rows, row stride = stride_elems.
// pad_interval_enc: 0=2,1=4,2=8,3=16,4=32,5=64 DWORDs between pads;
// pad_amount_enc:   0=1,1=2,... DWORDs of padding inserted.
// ---------------------------------------------------------------------------
__device__ __forceinline__ void tdm_load_2d(uint32_t lds_addr, const float* gptr,
                                            int tile_w, int tile_h, int stride_elems,
                                            int pad_interval_enc, int pad_amount_enc) {
    const uint64_t ga  = (uint64_t)(uintptr_t)gptr;
    const uint32_t td0 = (uint32_t)tile_w;     // tensor_dim0 == tile width (in-bounds tiles)
    const uint32_t td1 = (uint32_t)tile_h;
    const uint64_t st0 = (uint64_t)(uint32_t)stride_elems;

    v4u g0;
    g0[0] = 1u;                                            // count=1 (valid user D#)
    g0[1] = lds_addr;                                      // LDS byte address
    g0[2] = (uint32_t)ga;                                  // global_addr[31:0]
    g0[3] = (uint32_t)((ga >> 32) & 0x1FFFFFFu) | (2u << 30); // addr[56:32] | type=2

    v8i g1;
    g1[0] = (int)((2u << 16) |                             // data_size = 4 bytes
                  (1u << 20) |                             // pad_enable
                  ((uint32_t)pad_interval_enc << 22) |
                  ((uint32_t)pad_amount_enc << 25));
    g1[1] = (int)((td0 & 0xFFFFu) << 16);                  // tensor_dim0[15:0]
    g1[2] = (int)((td0 >> 16) | ((td1 & 0xFFFFu) << 16));  // tensor_dim0[31:16] | dim1[15:0]
    g1[3] = (int)((td1 >> 16) | ((uint32_t)tile_w << 16)); // dim1[31:16] | tile_dim0
    g1[4] = (int)(uint32_t)tile_h;                         // tile_dim1 (tile_dim2 = 0)
    g1[5] = (int)(uint32_t)(st0 & 0xFFFFFFFFu);            // tensor_dim0_stride[31:0]
    g1[6] = (int)(uint32_t)(st0 >> 32);                    // stride[47:32] (dim1_stride=0)
    g1[7] = 0;

    const v4i z4 = {0, 0, 0, 0};
#if defined(__clang_major__) && (__clang_major__ >= 23)
    const v8i z8 = {0, 0, 0, 0, 0, 0, 0, 0};
    __builtin_amdgcn_tensor_load_to_lds(g0, g1, z4, z4, z8, 0);
#else
    __builtin_amdgcn_tensor_load_to_lds(g0, g1, z4, z4, 0);
#endif
}

// ---------------------------------------------------------------------------
// LDS-tiled fp32 GEMM on the WMMA pipe, TDM-fed, double-buffered.
// C[M,N] = A[M,K] @ B[K,N], row-major. Block = 256 threads (8 waves).
// Block tile 64x64, K-chunk 32. Wave (wm,wn) in 2x4 grid owns a 32x16 slab
// (two 16x16 accumulators). LDS strides 34 / 66 (via TDM padding) kill bank
// conflicts on the fragment reads.
// ---------------------------------------------------------------------------
#define KC     32
#define LDA_S  34   // 32 + 2 pad (pad_interval enc 4 = 32 dw, amount enc 1 = 2 dw)
#define LDB_S  66   // 64 + 2 pad (pad_interval enc 5 = 64 dw, amount enc 1 = 2 dw)

__global__ void wmma_gemm_f32_tdm(const float* __restrict__ A,
                                  const float* __restrict__ Bm,
                                  float* __restrict__ C,
                                  int K, int lda, int ldb, int ldc) {
    __shared__ float As[2][64 * LDA_S];
    __shared__ float Bs[2][KC * LDB_S];

    const int tid  = threadIdx.x;        // 0..255
    const int lane = tid & 31;
    const int wave = tid >> 5;           // 0..7
    const int wm   = wave >> 2;          // 0..1 : 32-row slab
    const int wn   = wave & 3;           // 0..3 : 16-col slab
    const int m    = lane & 15;
    const int hw   = lane >> 4;          // half-wave select

    const int Mblk = blockIdx.y * 64;
    const int Nblk = blockIdx.x * 64;

    const uint32_t lds_a0 = (uint32_t)(uintptr_t)(void*)&As[0][0];
    const uint32_t lds_a1 = (uint32_t)(uintptr_t)(void*)&As[1][0];
    const uint32_t lds_b0 = (uint32_t)(uintptr_t)(void*)&Bs[0][0];
    const uint32_t lds_b1 = (uint32_t)(uintptr_t)(void*)&Bs[1][0];

    const float* Ablk = A  + (size_t)Mblk * lda;

    v8f acc0 = {}, acc1 = {};

    // Preload chunk 0 into buffer 0 (wave 0 drives the TDM; TENSORcnt is per-wave)
    if (wave == 0) {
        tdm_load_2d(lds_a0, Ablk, KC, 64, lda, 4, 1);
        tdm_load_2d(lds_b0, Bm + Nblk, 64, KC, ldb, 5, 1);
        __builtin_amdgcn_s_wait_tensorcnt(0);
    }
    __syncthreads();

    int buf = 0;
    for (int kc = 0; kc < K; kc += KC) {
        // Kick off next chunk into the other buffer while this one is consumed
        if (wave == 0 && (kc + KC) < K) {
            const uint32_t la = buf ? lds_a0 : lds_a1;
            const uint32_t lb = buf ? lds_b0 : lds_b1;
            tdm_load_2d(la, Ablk + kc + KC, KC, 64, lda, 4, 1);
            tdm_load_2d(lb, Bm + (size_t)(kc + KC) * ldb + Nblk, 64, KC, ldb, 5, 1);
        }

        const float* ar0 = &As[buf][(wm * 32 + m) * LDA_S];
        const float* ar1 = ar0 + 16 * LDA_S;
        const float* bc  = &Bs[buf][wn * 16 + m];
#pragma unroll
        for (int k0 = 0; k0 < KC; k0 += 4) {
            const int kl = k0 + 2 * hw;
            const v2f a0 = *(const v2f*)(ar0 + kl);
            const v2f a1 = *(const v2f*)(ar1 + kl);
            v2f b;
            b[0] = bc[kl * LDB_S];
            b[1] = bc[(kl + 1) * LDB_S];
            acc0 = __builtin_amdgcn_wmma_f32_16x16x4_f32(false, a0, false, b,
                                                         (short)0, acc0, false, false);
            acc1 = __builtin_amdgcn_wmma_f32_16x16x4_f32(false, a1, false, b,
                                                         (short)0, acc1, false, false);
        }

        if (wave == 0) __builtin_amdgcn_s_wait_tensorcnt(0);
        __syncthreads();
        buf ^= 1;
    }

    const int col = Nblk + wn * 16 + m;
#pragma unroll
    for (int v = 0; v < 8; ++v) {
        const int r0 = Mblk + wm * 32 + v + 8 * hw;
        C[(size_t)r0 * ldc + col]        = acc0[v];
        C[(size_t)(r0 + 16) * ldc + col] = acc1[v];
    }
}

// ---------------------------------------------------------------------------
// Simple per-wave WMMA GEMM (kept for the two small projections).
// mode 0: plain store.  mode 1: C = softplus(acc + bias[col]).
// Grid: x = Ntiles/2, y = Mtiles. Block: (32,2).
// ---------------------------------------------------------------------------
__global__ void wmma_gemm_f32(const float* __restrict__ A,
                              const float* __restrict__ Bm,
                              float* __restrict__ C,
                              int K, int lda, int ldb, int ldc,
                              const float* __restrict__ bias, int mode) {
    const int lane  = threadIdx.x;
    const int tileN = blockIdx.x * 2 + threadIdx.y;
    const int tileM = blockIdx.y;
    const int M0 = tileM * 16, N0 = tileN * 16;
    const int m  = lane & 15;
    const int hw = lane >> 4;

    v8f acc = {};
    const float* arow = A  + (size_t)(M0 + m) * lda;
    const float* bcol = Bm + (size_t)(N0 + m);
    for (int k0 = 0; k0 < K; k0 += 4) {
        const int kl = k0 + 2 * hw;
        v2f a, b;
        a[0] = arow[kl];
        a[1] = arow[kl + 1];
        b[0] = bcol[(size_t)kl * ldb];
        b[1] = bcol[(size_t)(kl + 1) * ldb];
        acc = __builtin_amdgcn_wmma_f32_16x16x4_f32(false, a, false, b,
                                                    (short)0, acc, false, false);
    }

    const int col = N0 + m;
    const float bv = (mode == 1) ? bias[col] : 0.0f;
#pragma unroll
    for (int v = 0; v < 8; ++v) {
        const int row = M0 + v + 8 * hw;
        float r = acc[v];
        if (mode == 1) r = softplus_f(r + bv);
        C[(size_t)row * ldc + col] = r;
    }
}

// ---------------------------------------------------------------------------
// Depthwise conv (k=4, pad (1,2)) + exact GELU over both branches.
// ---------------------------------------------------------------------------
__global__ void dwconv_gelu(const float* __restrict__ xz,   // (B,L,2048)
                            const float* __restrict__ Kx, const float* __restrict__ bx,
                            const float* __restrict__ Kz, const float* __restrict__ bz,
                            float* __restrict__ xg,         // (B,L,1024)
                            float* __restrict__ ycat) {     // (B,L,2048)
    const int c = blockIdx.x * blockDim.x + threadIdx.x;    // 0..2047
    const int b = blockIdx.y;
    float k0, k1, k2, k3, bias;
    if (c < DH) {
        k0 = Kx[c*4+0]; k1 = Kx[c*4+1]; k2 = Kx[c*4+2]; k3 = Kx[c*4+3];
        bias = bx[c];
    } else {
        const int cz = c - DH;
        k0 = Kz[cz*4+0]; k1 = Kz[cz*4+1]; k2 = Kz[cz*4+2]; k3 = Kz[cz*4+3];
        bias = bz[cz];
    }
    const float* xp = xz + (size_t)b * L_ * DI + c;
    float xm1 = 0.0f;
    float x0 = xp[0];
    float x1 = xp[(size_t)1 * DI];
    float x2 = xp[(size_t)2 * DI];
    for (int l = 0; l < L_; ++l) {
        float y = gelu_f(xm1 * k0 + x0 * k1 + x1 * k2 + x2 * k3 + bias);
        if (c < DH) xg  [((size_t)b * L_ + l) * DH + c] = y;
        else        ycat[((size_t)b * L_ + l) * DI + c] = y;
        xm1 = x0; x0 = x1; x1 = x2;
        x2 = (l + 3 < L_) ? xp[(size_t)(l + 3) * DI] : 0.0f;
    }
}

// ---------------------------------------------------------------------------
// Selective scan: thread per (b,d); h[16] in registers; 1024 sequential steps.
// ---------------------------------------------------------------------------
__global__ void selective_scan(const float* __restrict__ delta, // (B,L,DH), softplus'd
                               const float* __restrict__ u,     // (B,L,DH)
                               const float* __restrict__ xdbl,  // (B,L,96)
                               const float* __restrict__ A_log, // (DH,16)
                               const float* __restrict__ Dp,    // (DH)
                               float* __restrict__ ycat) {      // (B,L,2048)
    const int d = blockIdx.x * blockDim.x + threadIdx.x;
    const int b = blockIdx.y;
    float Ad[DS], h[DS];
#pragma unroll
    for (int n = 0; n < DS; ++n) {
        Ad[n] = -__expf(A_log[(size_t)d * DS + n]);
        h[n] = 0.0f;
    }
    const float Dd = Dp[d];
    const float* dl = delta + (size_t)b * L_ * DH + d;
    const float* ul = u     + (size_t)b * L_ * DH + d;
    const float* xb = xdbl  + (size_t)b * L_ * NPROJ;
    float* yo = ycat + (size_t)b * L_ * DI + d;
    for (int l = 0; l < L_; ++l) {
        const float dt = dl[(size_t)l * DH];
        const float uu = ul[(size_t)l * DH];
        const float du = dt * uu;
        const float* bc = xb + (size_t)l * NPROJ + DTR;
        float y = 0.0f;
#pragma unroll
        for (int n = 0; n < DS; ++n) {
            const float hn = __expf(dt * Ad[n]) * h[n] + du * bc[n];
            h[n] = hn;
            y += hn * bc[DS + n];
        }
        yo[(size_t)l * DI] = y + uu * Dd;
    }
}

// ---------------------------------------------------------------------------
extern "C" void kernel_launch(void* const* d_in, const int* in_sizes, int n_in,
                              void* d_out, int out_size, void* d_ws, size_t ws_size,
                              hipStream_t stream) {
    (void)in_sizes; (void)n_in; (void)out_size; (void)ws_size;
    const float* hs      = (const float*)d_in[0];
    const float* W_in    = (const float*)d_in[1];
    const float* W_xproj = (const float*)d_in[2];
    const float* W_dt    = (const float*)d_in[3];
    const float* b_dt    = (const float*)d_in[4];
    const float* A_log   = (const float*)d_in[5];
    const float* Dvec    = (const float*)d_in[6];
    const float* W_out   = (const float*)d_in[7];
    const float* K_x     = (const float*)d_in[8];
    const float* b_x     = (const float*)d_in[9];
    const float* K_z     = (const float*)d_in[10];
    const float* b_z     = (const float*)d_in[11];
    float* out = (float*)d_out;

    float* ws = (float*)d_ws;
    size_t o = 0;
    float* xz   = ws + o; o += (size_t)B_ * L_ * DI;     // in_proj out  (b,l,2048)
    float* xg   = ws + o; o += (size_t)B_ * L_ * DH;     // conv+gelu x  (b,l,1024)
    float* xdbl = ws + o; o += (size_t)B_ * L_ * NPROJ;  // x_proj out   (b,l,96)
    float* delt = ws + o; o += (size_t)B_ * L_ * DH;     // softplus(dt) (b,l,1024)
    float* ycat = ws + o; o += (size_t)B_ * L_ * DI;     // [y | z]      (b,l,2048)

    const int Mt = (B_ * L_) / 16;     // 256 row tiles for the small GEMMs

    // 1) in_proj: xz = hs @ W_in        (4096 x 2048, K=1024)  [TDM + LDS tiled]
    wmma_gemm_f32_tdm<<<dim3(DI / 64, (B_ * L_) / 64), 256, 0, stream>>>(
        hs, W_in, xz, DM, DM, DI, DI);

    // 2) depthwise conv + GELU on both branches
    dwconv_gelu<<<dim3(DI / 256, B_), 256, 0, stream>>>(
        xz, K_x, b_x, K_z, b_z, xg, ycat);

    // 3) x_proj: xdbl = xg @ W_xproj    (4096 x 96, K=1024)
    wmma_gemm_f32<<<dim3(NPROJ / 32, Mt), dim3(32, 2), 0, stream>>>(
        xg, W_xproj, xdbl, DH, DH, NPROJ, NPROJ, nullptr, 0);

    // 4) dt proj + bias + softplus: delt = sp(xdbl[:, :64] @ W_dt + b_dt)
    wmma_gemm_f32<<<dim3(DH / 32, Mt), dim3(32, 2), 0, stream>>>(
        xdbl, W_dt, delt, DTR, NPROJ, DH, DH, b_dt, 1);

    // 5) selective scan -> ycat[:,:,0:1024]
    selective_scan<<<dim3(DH / 256, B_), 256, 0, stream>>>(
        delt, xg, xdbl, A_log, Dvec, ycat);

    // 6) out_proj: out = ycat @ W_out   (4096 x 1024, K=2048)  [TDM + LDS tiled]
    wmma_gemm_f32_tdm<<<dim3(DM / 64, (B_ * L_) / 64), 256, 0, stream>>>(
        ycat, W_out, out, DI, DI, DM, DM);
}